// Net_9397388443957
// MI455X (gfx1250) — compile-verified
//
#include <hip/hip_runtime.h>

// GCN on MI455X (gfx1250, wave32). f32 WMMA (16x16x4) for both GEMMs;
// edge aggregation via L2-resident f32 atomics (agg arrays are 6.4 MB << 192 MB L2).
//
// WMMA k-slot permutation trick: the hardware reduces over 4 K-slots per
// instruction; A slot (lane-half, vgpr) pairs with B slot (lane-half, vgpr).
// Any bijective assignment of physical k's to slots is valid as long as A and
// B use the same assignment. We assign half h the k's {k0+4h .. k0+4h+3} so
// each lane's A fetch is one aligned global_load_b128.

typedef __attribute__((ext_vector_type(2))) float v2f;
typedef __attribute__((ext_vector_type(4))) float v4f;
typedef __attribute__((ext_vector_type(8))) float v8f;

#define DIN  512
#define HID  16   // layer-1 width == WMMA N tile
#define DOUT 10

// ---------------- degree / normalization ----------------
__global__ void k_deg_init(float* __restrict__ deg, int n) {
    int i = blockIdx.x * blockDim.x + threadIdx.x;
    if (i < n) deg[i] = 1.0f;  // self-loop
}

__global__ void k_deg_count(const long long* __restrict__ dst, float* __restrict__ deg, int e) {
    int i = blockIdx.x * blockDim.x + threadIdx.x;
    if (i < e) atomicAdd(&deg[(int)dst[i]], 1.0f);
}

__global__ void k_dinv(float* __restrict__ deg, int n) {
    int i = blockIdx.x * blockDim.x + threadIdx.x;
    if (i < n) deg[i] = rsqrtf(deg[i]);  // deg >= 1 always (self-loop)
}

// ---------------- GEMM1: h1 = x @ W1  (N x 512) @ (512 x 16) ----------------
// One wave per 16-row tile: 64 global_load_b128 + 128 v_wmma_f32_16x16x4_f32.
__global__ void __launch_bounds__(256) k_gemm1(const float* __restrict__ x,
                                               const float* __restrict__ W1,
                                               float* __restrict__ h1,
                                               int ntiles) {
    __shared__ float w[DIN * HID];                 // 32 KB of LDS
    for (int i = threadIdx.x; i < DIN * HID; i += 256) w[i] = W1[i];
    __syncthreads();

    const int wave = threadIdx.x >> 5;             // 8 waves / block
    const int lane = threadIdx.x & 31;
    const int tile = blockIdx.x * 8 + wave;
    if (tile >= ntiles) return;                    // wave-uniform: EXEC all-ones inside

    const int half = lane >> 4;                    // lane-half picks its 4 k's
    const int l16  = lane & 15;
    const int row  = tile * 16 + l16;              // A-matrix M index
    const float* __restrict__ xr = x + (size_t)row * DIN;

    v8f c = {0.f, 0.f, 0.f, 0.f, 0.f, 0.f, 0.f, 0.f};
    #pragma unroll 4
    for (int k0 = 0; k0 < DIN; k0 += 8) {
        const int ka = k0 + 4 * half;              // this lane's 4 contiguous k's
        v4f av = *(const v4f*)(xr + ka);           // one 16B-aligned b128 load
        v2f a0 = {av.x, av.y};                     // slots for WMMA #1
        v2f a1 = {av.z, av.w};                     // slots for WMMA #2
        v2f b0, b1;                                // B rows matching the same k's
        b0.x = w[(ka + 0) * HID + l16];
        b0.y = w[(ka + 1) * HID + l16];
        b1.x = w[(ka + 2) * HID + l16];
        b1.y = w[(ka + 3) * HID + l16];
        c = __builtin_amdgcn_wmma_f32_16x16x4_f32(false, a0, false, b0,
                                                  (short)0, c, false, false);
        c = __builtin_amdgcn_wmma_f32_16x16x4_f32(false, a1, false, b1,
                                                  (short)0, c, false, false);
    }

    // C/D layout: VGPR i -> M = i + 8*half, N = l16
    const int rbase = tile * 16 + 8 * half;
    #pragma unroll
    for (int i = 0; i < 8; ++i)
        h1[(size_t)(rbase + i) * HID + l16] = c[i];
}

// ---------------- aggregation (16-wide feature arrays) ----------------
// agg[n][f] = h[n][f] * dinv[n]^2   (self-loop message)
__global__ void k_agg_self(const float* __restrict__ h, const float* __restrict__ dinv,
                           float* __restrict__ agg, int n) {
    int t = blockIdx.x * blockDim.x + threadIdx.x;
    if (t < n * HID) {
        int node = t >> 4;
        float di = dinv[node];
        agg[t] = h[t] * di * di;
    }
}

// agg[dst][:] += h[src][:] * dinv[src]*dinv[dst]   one thread per edge
__global__ void k_agg_edges(const long long* __restrict__ src, const long long* __restrict__ dst,
                            const float* __restrict__ h, const float* __restrict__ dinv,
                            float* __restrict__ agg, int e) {
    int i = blockIdx.x * blockDim.x + threadIdx.x;
    if (i < e) {
        int s = (int)src[i];
        int d = (int)dst[i];
        float nrm = dinv[s] * dinv[d];
        const v4f* __restrict__ hs = (const v4f*)(h + (size_t)s * HID);
        float* __restrict__ ad = agg + (size_t)d * HID;
        #pragma unroll
        for (int j = 0; j < 4; ++j) {
            v4f m = hs[j];                         // global_load_b128 gather
            atomicAdd(ad + 4 * j + 0, m.x * nrm);  // non-returning f32 atomics, L2-resident
            atomicAdd(ad + 4 * j + 1, m.y * nrm);
            atomicAdd(ad + 4 * j + 2, m.z * nrm);
            atomicAdd(ad + 4 * j + 3, m.w * nrm);
        }
    }
}

// ---------------- GEMM2: h2 = relu(agg1 + b1) @ W2pad  (N x 16) @ (16 x 16) ----------
__global__ void __launch_bounds__(256) k_gemm2(const float* __restrict__ agg1,
                                               const float* __restrict__ b1,
                                               const float* __restrict__ W2,
                                               float* __restrict__ h2,
                                               int ntiles) {
    __shared__ float w[16 * 16];
    __shared__ float bias[16];
    {
        int k = threadIdx.x >> 4, nn = threadIdx.x & 15;
        if (threadIdx.x < 256) w[threadIdx.x] = (nn < DOUT) ? W2[k * DOUT + nn] : 0.0f;
        if (threadIdx.x < 16) bias[threadIdx.x] = b1[threadIdx.x];
    }
    __syncthreads();

    const int wave = threadIdx.x >> 5;
    const int lane = threadIdx.x & 31;
    const int tile = blockIdx.x * 8 + wave;
    if (tile >= ntiles) return;

    const int half = lane >> 4;
    const int l16  = lane & 15;
    const int row  = tile * 16 + l16;
    const float* __restrict__ ar = agg1 + (size_t)row * HID;

    v8f c = {0.f, 0.f, 0.f, 0.f, 0.f, 0.f, 0.f, 0.f};
    #pragma unroll
    for (int k0 = 0; k0 < 16; k0 += 8) {
        const int ka = k0 + 4 * half;
        v4f av = *(const v4f*)(ar + ka);           // b128 load of this lane's k's
        av.x = fmaxf(av.x + bias[ka + 0], 0.0f);   // fused bias + ReLU
        av.y = fmaxf(av.y + bias[ka + 1], 0.0f);
        av.z = fmaxf(av.z + bias[ka + 2], 0.0f);
        av.w = fmaxf(av.w + bias[ka + 3], 0.0f);
        v2f a0 = {av.x, av.y};
        v2f a1 = {av.z, av.w};
        v2f b0, b1;
        b0.x = w[(ka + 0) * 16 + l16];
        b0.y = w[(ka + 1) * 16 + l16];
        b1.x = w[(ka + 2) * 16 + l16];
        b1.y = w[(ka + 3) * 16 + l16];
        c = __builtin_amdgcn_wmma_f32_16x16x4_f32(false, a0, false, b0,
                                                  (short)0, c, false, false);
        c = __builtin_amdgcn_wmma_f32_16x16x4_f32(false, a1, false, b1,
                                                  (short)0, c, false, false);
    }

    const int rbase = tile * 16 + 8 * half;
    #pragma unroll
    for (int i = 0; i < 8; ++i)
        h2[(size_t)(rbase + i) * HID + l16] = c[i];   // cols 10..15 are exact zeros
}

// ---------------- final: out = log_softmax(agg2 + b2) over 10 classes ------------
__global__ void k_logsoftmax(const float* __restrict__ agg2, const float* __restrict__ b2,
                             float* __restrict__ out, int n) {
    int i = blockIdx.x * blockDim.x + threadIdx.x;
    if (i < n) {
        float v[DOUT];
        float m = -__builtin_inff();
        #pragma unroll
        for (int f = 0; f < DOUT; ++f) {
            v[f] = agg2[(size_t)i * HID + f] + b2[f];
            m = fmaxf(m, v[f]);
        }
        float s = 0.0f;
        #pragma unroll
        for (int f = 0; f < DOUT; ++f) s += __expf(v[f] - m);
        float l = __logf(s);
        #pragma unroll
        for (int f = 0; f < DOUT; ++f) out[(size_t)i * DOUT + f] = v[f] - m - l;
    }
}

// ---------------- host-side orchestration ----------------
extern "C" void kernel_launch(void* const* d_in, const int* in_sizes, int n_in,
                              void* d_out, int out_size, void* d_ws, size_t ws_size,
                              hipStream_t stream) {
    const float*     x  = (const float*)d_in[0];
    const long long* ei = (const long long*)d_in[1];
    const float*     W1 = (const float*)d_in[2];
    const float*     b1 = (const float*)d_in[3];
    const float*     W2 = (const float*)d_in[4];
    const float*     b2 = (const float*)d_in[5];
    float*           out = (float*)d_out;

    const int N = in_sizes[0] / DIN;    // 100000 (multiple of 16)
    const int E = in_sizes[1] / 2;      // 3200000
    const long long* src = ei;
    const long long* dst = ei + E;

    // workspace layout (all f32, re-initialized every call; ~26 MB)
    float* ws   = (float*)d_ws;
    float* dinv = ws;                                 // N      (deg -> dinv in place)
    float* h1   = dinv + N;                           // N*16
    float* agg1 = h1   + (size_t)N * HID;             // N*16
    float* h2   = agg1 + (size_t)N * HID;             // N*16
    float* agg2 = h2   + (size_t)N * HID;             // N*16

    const int ntiles = N / 16;

    k_deg_init <<<(N + 255) / 256, 256, 0, stream>>>(dinv, N);
    k_deg_count<<<(E + 255) / 256, 256, 0, stream>>>(dst, dinv, E);
    k_dinv     <<<(N + 255) / 256, 256, 0, stream>>>(dinv, N);

    k_gemm1    <<<(ntiles + 7) / 8, 256, 0, stream>>>(x, W1, h1, ntiles);
    k_agg_self <<<(N * HID + 255) / 256, 256, 0, stream>>>(h1, dinv, agg1, N);
    k_agg_edges<<<(E + 255) / 256, 256, 0, stream>>>(src, dst, h1, dinv, agg1, E);

    k_gemm2    <<<(ntiles + 7) / 8, 256, 0, stream>>>(agg1, b1, W2, h2, ntiles);
    k_agg_self <<<(N * HID + 255) / 256, 256, 0, stream>>>(h2, dinv, agg2, N);
    k_agg_edges<<<(E + 255) / 256, 256, 0, stream>>>(src, dst, h2, dinv, agg2, E);

    k_logsoftmax<<<(N + 255) / 256, 256, 0, stream>>>(agg2, b2, out, N);
}